// LosslessPool_32804960207046
// MI455X (gfx1250) — compile-verified
//
#include <hip/hip_runtime.h>
#include <stdint.h>

// Space-to-depth (k=2) on [32,224,224,64] f32 -> [32,112,112,256].
// Pure permutation: 822 MB total traffic -> ~35us floor at 23.3 TB/s.
// Entirely on the CDNA5 async global<->LDS DMA path (ASYNCcnt):
//   global_load_async_to_lds_b128 / global_store_async_from_lds_b128
// One 16 KB stage (16 output pixels, never crosses an output row since
// 112 % 16 == 0) per 256-thread block, mapped on a 3D grid (7,112,32) so the
// stage decomposition needs zero divisions. Latency is hidden by occupancy
// (16 KB LDS/WG -> ~20 WGs per WGP). No barriers: each thread stores exactly
// the LDS bytes it loaded. Per-u strides (+2048B on both global and LDS side)
// live in the instruction's 24-bit offset immediate, which the ISA adds to
// BOTH addresses. Non-temporal hints: every byte touched exactly once.

#define ASYNC_LOAD_B128(lds, voff, sbase, K)                                   \
  asm volatile("global_load_async_to_lds_b128 %0, %1, %2 offset:" #K           \
               " th:TH_LOAD_NT"                                                \
               :: "v"(lds), "v"(voff), "s"(sbase) : "memory")

#define ASYNC_STORE_B128(voff, lds, sbase, K)                                  \
  asm volatile("global_store_async_from_lds_b128 %0, %1, %2 offset:" #K        \
               " th:TH_STORE_NT"                                               \
               :: "v"(voff), "v"(lds), "s"(sbase) : "memory")

#define WAIT_ASYNC0() asm volatile("s_wait_asynccnt 0" ::: "memory")

__global__ __launch_bounds__(256) void s2d_async_kernel(
    const float* __restrict__ src, float* __restrict__ dst) {
  // 16 KB: kh*8192 + pix*512 + c4*16, pix 0..15
  __shared__ float4 smem[1024];

  const uint32_t tid = threadIdx.x;
  const uint32_t c4  = tid & 31u;   // float4 within a 512B chunk
  const uint32_t jt  = tid >> 5;    // 0..7
  const uint32_t kh  = jt & 1u;     // input row select
  const uint32_t pt  = jt >> 1;     // 0..3

  const uint32_t v_lds = ((uint32_t)(uintptr_t)&smem[0])
                         + kh * 8192u + pt * 512u + c4 * 16u;
  const uint32_t v_in  = kh * 57344u + pt * 512u + c4 * 16u; // 57344 = 224*64*4
  const uint32_t v_out = tid << 4;

  // Stage coordinates straight from the grid: no division anywhere.
  const uint32_t x = blockIdx.x;    // 0..6   : 16-pixel tile within output row
  const uint32_t y = blockIdx.y;    // 0..111 : oh
  const uint32_t z = blockIdx.z;    // 0..31  : batch
  // Shared term: z*224*224*256 == z*112*112*1024, y*2*224*256 == y*7*16384.
  const uint32_t common = z * 12845056u + y * 114688u;
  const char* sload  = (const char*)src + (common + x * 8192u);
  char*       sstore = (char*)dst + (common + x * 16384u);

  // 4 async loads: per-u stride entirely in the offset immediate.
  ASYNC_LOAD_B128(v_lds, v_in, sload, 0);
  ASYNC_LOAD_B128(v_lds, v_in, sload, 2048);
  ASYNC_LOAD_B128(v_lds, v_in, sload, 4096);
  ASYNC_LOAD_B128(v_lds, v_in, sload, 6144);

  WAIT_ASYNC0();  // this wave's loads landed in LDS

  // 4 async stores: offset feeds the LDS side (+2048/u); the extra +2048/u
  // for the global side is folded into the uniform SGPR base.
  ASYNC_STORE_B128(v_out, v_lds, sstore,        0);
  ASYNC_STORE_B128(v_out, v_lds, sstore + 2048, 2048);
  ASYNC_STORE_B128(v_out, v_lds, sstore + 4096, 4096);
  ASYNC_STORE_B128(v_out, v_lds, sstore + 6144, 6144);

  WAIT_ASYNC0();  // defensive; s_endpgm also implies wait-idle
}

extern "C" void kernel_launch(void* const* d_in, const int* in_sizes, int n_in,
                              void* d_out, int out_size, void* d_ws, size_t ws_size,
                              hipStream_t stream) {
  (void)in_sizes; (void)n_in; (void)d_ws; (void)ws_size; (void)out_size;
  const float* src = (const float*)d_in[0];
  float*       dst = (float*)d_out;

  // [32,112,112,256] output = 25,088 stages of 16 KB = grid (7, 112, 32).
  dim3 grid(7, 112, 32);
  s2d_async_kernel<<<grid, 256, 0, stream>>>(src, dst);
}